// FastWeightsState_82094004896067
// MI455X (gfx1250) — compile-verified
//
#include <hip/hip_runtime.h>

typedef __attribute__((ext_vector_type(2))) float v2f;
typedef __attribute__((ext_vector_type(8))) float v8f;

#define DIM        4096
#define ROWS_PB    16           // rows per block (one 16-row WMMA stripe)
#define THREADS    256          // 8 waves of 32

// out = x + (x @ U) @ V^T, rank-4 fast-weight update, f32 throughout.
// Memory-bound: ideal traffic = read x once (134 MB) + write out (134 MB)
// ~= 11.5 us at 23.3 TB/s. Stage 1 streams x (RT, re-read soon by stage 3);
// stage 3 re-reads x from WGP$/L2 with NT (last use) and NT-stores out.
__global__ __launch_bounds__(THREADS) void fastweights_rank4_kernel(
    const float* __restrict__ x,
    const float* __restrict__ U,   // [DIM][4] row-major
    const float* __restrict__ V,   // [DIM][4] row-major
    float* __restrict__ out)
{
    __shared__ float4 ldsU[DIM];            // U[k][0..3]          : 64 KB
    __shared__ float  partial[16][16][4];   // per-(row,sub) P part:  4 KB
    __shared__ float  Pf[16][4];            // reduced P stripe

    const int t = threadIdx.x;
    const long rowBase = (long)blockIdx.x * ROWS_PB;

    // ---- Stage 0: U -> LDS (float4 per k; U rows are contiguous) ----
    {
        const float4* U4 = (const float4*)U;
        for (int k = t; k < DIM; k += THREADS) ldsU[k] = U4[k];
    }
    __syncthreads();

    // ---- Stage 1: partial P = x @ U for this 16-row stripe ----
    // thread t: row r = t>>4, sub s = t&15; 16 threads per row, each does 64
    // float4 loads strided so the 16 subs cover a contiguous 256B span.
    const int r = t >> 4;
    const int s = t & 15;
    {
        const float* xrow = x + (rowBase + r) * (long)DIM;
        float a0 = 0.f, a1 = 0.f, a2 = 0.f, a3 = 0.f;
        #pragma unroll 4
        for (int j = 0; j < 64; ++j) {
            const int c = (s << 2) + (j << 6);
            const float4 xv = *(const float4*)(xrow + c);
            const float4 u0 = ldsU[c + 0];
            const float4 u1 = ldsU[c + 1];
            const float4 u2 = ldsU[c + 2];
            const float4 u3 = ldsU[c + 3];
            a0 += xv.x * u0.x + xv.y * u1.x + xv.z * u2.x + xv.w * u3.x;
            a1 += xv.x * u0.y + xv.y * u1.y + xv.z * u2.y + xv.w * u3.y;
            a2 += xv.x * u0.z + xv.y * u1.z + xv.z * u2.z + xv.w * u3.z;
            a3 += xv.x * u0.w + xv.y * u1.w + xv.z * u2.w + xv.w * u3.w;
        }
        partial[r][s][0] = a0;
        partial[r][s][1] = a1;
        partial[r][s][2] = a2;
        partial[r][s][3] = a3;
    }
    __syncthreads();

    // ---- Stage 2: reduce 16 subs -> Pf[16][4] ----
    if (t < 64) {
        const int rr = t >> 2;
        const int cc = t & 3;
        float sum = 0.f;
        #pragma unroll
        for (int k = 0; k < 16; ++k) sum += partial[rr][k][cc];
        Pf[rr][cc] = sum;
    }
    __syncthreads();

    // ---- Stage 3: out tile = x tile + P(16x4) x V^T(4x16) via WMMA f32 16x16x4 ----
    // A layout (32-bit A 16x4, ISA 7.12.2): lanes 0-15 -> M=lane, VGPR0=K0, VGPR1=K1;
    //                                       lanes 16-31 -> M=lane-16, VGPR0=K2, VGPR1=K3.
    // B layout (4x16): VGPR0 = {K0 lanes0-15, K2 lanes16-31}, VGPR1 = {K1, K3};
    //                  B[k][n] = V[col0+n][k] -> one float4 V-row load per lane.
    const int wave = t >> 5;
    const int lane = t & 31;
    const int lr   = lane & 15;
    const bool hi  = lane >= 16;

    v2f amat;
    amat.x = Pf[lr][hi ? 2 : 0];
    amat.y = Pf[lr][hi ? 3 : 1];

    const float4* V4 = (const float4*)V;
    const long mBase = rowBase + (hi ? 8 : 0);   // C/D: VGPR v holds M = v (+8 for hi lanes)

    for (int nt = wave; nt < DIM / 16; nt += 8) {
        const int col0 = nt << 4;

        const float4 vv = V4[col0 + lr];
        v2f bmat;
        bmat.x = hi ? vv.z : vv.x;
        bmat.y = hi ? vv.w : vv.y;

        // Load C = x tile: last use of these x lines -> non-temporal hint so
        // WGP$/L2 retention is reserved for the not-yet-consumed stream.
        const float* xp = x + mBase * (long)DIM + col0 + lr;
        v8f cmat;
        #pragma unroll
        for (int v = 0; v < 8; ++v)
            cmat[v] = __builtin_nontemporal_load(xp + (long)v * DIM);

        // D = A x B + C   (8 args: neg_a, A, neg_b, B, c_mod, C, reuse_a, reuse_b)
        v8f dmat = __builtin_amdgcn_wmma_f32_16x16x4_f32(
            false, amat, false, bmat, (short)0, cmat, false, false);

        // out is write-once and never re-read: non-temporal stores keep the
        // 134 MB output stream from evicting x in WGP$/L2.
        float* op = out + mBase * (long)DIM + col0 + lr;
        #pragma unroll
        for (int v = 0; v < 8; ++v)
            __builtin_nontemporal_store(dmat[v], op + (long)v * DIM);
    }
}

extern "C" void kernel_launch(void* const* d_in, const int* in_sizes, int n_in,
                              void* d_out, int out_size, void* d_ws, size_t ws_size,
                              hipStream_t stream) {
    const float* x = (const float*)d_in[0];   // [8192, 4096] f32
    const float* U = (const float*)d_in[1];   // [4096, 4]    f32
    const float* V = (const float*)d_in[2];   // [4096, 4]    f32
    float* out = (float*)d_out;               // [8192, 4096] f32

    const int nrows = in_sizes[0] / DIM;      // 8192
    const int grid  = nrows / ROWS_PB;        // 512 blocks

    fastweights_rank4_kernel<<<grid, THREADS, 0, stream>>>(x, U, V, out);
}